// RecurrentAdamclipAttention_22265110463162
// MI455X (gfx1250) — compile-verified
//
#include <hip/hip_runtime.h>

// RecurrentAdamclipAttention — MI455X (gfx1250), wave32.
// Memory-bound streaming kernel: 1 block per (n,h), fused Adam update,
// Si_new cached in LDS, V = z * (q^T @ Si_new) via V_WMMA_F32_16X16X4_F32.

typedef float v2f __attribute__((ext_vector_type(2)));
typedef float v8f __attribute__((ext_vector_type(8)));

constexpr float kMU   = 0.9f;
constexpr float kSTEP = 0.001f;
constexpr float kBETA = 0.999f;
constexpr float kEPS  = 1e-6f;

__device__ __forceinline__ float feat(float x) {
    // elu(x) + 1
    return x > 0.0f ? (x + 1.0f) : __expf(x);
}

__device__ __forceinline__ void adam_upd(float s, float p, float m, float u,
                                         float& pn, float& mn, float& sn) {
    float uc = fminf(fmaxf(u, -2.0f), 2.0f);
    pn = kMU * p - kSTEP * uc;
    mn = kBETA * m + (1.0f - kBETA) * (u * u);
    sn = s - pn / sqrtf(mn + 1e-16f);
}

__global__ __launch_bounds__(256) void ra_attn_kernel(
    const float* __restrict__ q,  const float* __restrict__ k,
    const float* __restrict__ val, const float* __restrict__ Si,
    const float* __restrict__ Zi, const float* __restrict__ Pi,
    const float* __restrict__ Mi,
    float* __restrict__ oV,  float* __restrict__ oSi, float* __restrict__ oZi,
    float* __restrict__ oPi, float* __restrict__ oMi, float* __restrict__ oUi)
{
    __shared__ __align__(16) float Ss[64 * 64];   // Si_new tile, row-major (d, m)
    __shared__ __align__(16) float Qs[64];        // feat(q)
    __shared__ __align__(16) float Ks[64];        // feat(k)
    __shared__ __align__(16) float Vs[64];        // value row
    __shared__ __align__(16) float Zs[64];        // Zi_new
    __shared__ float zinv_s;

    const int    b   = blockIdx.x;      // b = n*H + h
    const int    t   = threadIdx.x;     // 256 threads = 8 wave32
    const int    b64 = b * 64;
    const size_t bS  = (size_t)b * 4096;

    // --- stage small vectors + feature maps + Zi update -------------------
    if (t < 64) {
        float kf = feat(k[b64 + t]);
        float zn = Zi[b64 + t] + kf;
        oZi[b64 + t] = zn;
        Ks[t] = kf;
        Zs[t] = zn;
    } else if (t < 128) {
        Vs[t - 64] = val[b64 + (t - 64)];
    } else if (t < 192) {
        Qs[t - 128] = feat(q[b64 + (t - 128)]);
    }
    __syncthreads();

    // --- normalizer z = 1 / (Q . Zi_new + eps) ---------------------------
    if (t == 0) {
        float acc = 0.0f;
        #pragma unroll
        for (int d = 0; d < 64; ++d) acc += Qs[d] * Zs[d];
        zinv_s = 1.0f / (acc + kEPS);
    }

    // --- fused elementwise Adam update over the 64x64 state --------------
    // 4096 elems / 256 threads = 16 per thread = 4 float4, fully coalesced.
    const float4* Si4 = (const float4*)(Si + bS);
    const float4* Pi4 = (const float4*)(Pi + bS);
    const float4* Mi4 = (const float4*)(Mi + bS);
    float4* oSi4 = (float4*)(oSi + bS);
    float4* oPi4 = (float4*)(oPi + bS);
    float4* oMi4 = (float4*)(oMi + bS);
    float4* oUi4 = (float4*)(oUi + bS);

    #pragma unroll
    for (int j = 0; j < 4; ++j) {
        const int idx4 = j * 256 + t;       // float4 index within the 64x64 tile
        const int e    = idx4 * 4;          // element index
        const int d    = e >> 6;
        const int m    = e & 63;            // m % 4 == 0

        const float  kf = Ks[d];
        const float4 vv = *(const float4*)(&Vs[m]);
        const float4 s  = Si4[idx4];
        const float4 p  = Pi4[idx4];
        const float4 mm = Mi4[idx4];

        float4 u, pn, mn, sn;
        u.x = kf * vv.x; u.y = kf * vv.y; u.z = kf * vv.z; u.w = kf * vv.w;
        adam_upd(s.x, p.x, mm.x, u.x, pn.x, mn.x, sn.x);
        adam_upd(s.y, p.y, mm.y, u.y, pn.y, mn.y, sn.y);
        adam_upd(s.z, p.z, mm.z, u.z, pn.z, mn.z, sn.z);
        adam_upd(s.w, p.w, mm.w, u.w, pn.w, mn.w, sn.w);

        oUi4[idx4] = u;
        oPi4[idx4] = pn;
        oMi4[idx4] = mn;
        oSi4[idx4] = sn;
        *(float4*)(&Ss[e]) = sn;            // keep Si_new on-chip for the GEMV
    }
    __syncthreads();

    // --- V = z * (q^T @ Si_new) via V_WMMA_F32_16X16X4_F32 ---------------
    // A (16x4): q[k0..k0+3] replicated into every row -> every D row equals
    // the V segment. Lanes 0-15 carry K0/K1, lanes 16-31 carry K2/K3.
    // B (4x16): Si_new[k, m0+n], n striped over lanes within each half-wave.
    // All 8 waves execute the WMMA loop with full EXEC (waves 4-7 duplicate
    // waves 0-3); only the store is predicated.
    {
        const int wave = t >> 5;
        const int lane = t & 31;
        const int m0   = (wave & 3) * 16;   // 16-column tile of V
        const int hi   = lane >> 4;         // 0: K0/K1, 1: K2/K3
        const int col  = m0 + (lane & 15);

        v8f c = {0.0f, 0.0f, 0.0f, 0.0f, 0.0f, 0.0f, 0.0f, 0.0f};
        #pragma unroll
        for (int k0 = 0; k0 < 64; k0 += 4) {
            const int ka = k0 + 2 * hi;
            v2f a, bb;
            a.x  = Qs[ka];
            a.y  = Qs[ka + 1];
            bb.x = Ss[ka * 64 + col];
            bb.y = Ss[(ka + 1) * 64 + col];
            c = __builtin_amdgcn_wmma_f32_16x16x4_f32(
                    /*neg_a=*/false, a, /*neg_b=*/false, bb,
                    /*c_mod=*/(short)0, c, /*reuse_a=*/false, /*reuse_b=*/false);
        }
        // D-tile row 0 = VGPR0 on lanes 0-15 (N = lane).
        if (wave < 4 && lane < 16) {
            oV[b64 + col] = zinv_s * c[0];
        }
    }
}

extern "C" void kernel_launch(void* const* d_in, const int* in_sizes, int n_in,
                              void* d_out, int out_size, void* d_ws, size_t ws_size,
                              hipStream_t stream) {
    (void)n_in; (void)out_size; (void)d_ws; (void)ws_size;

    const float* q   = (const float*)d_in[0];  // (N,H,D)
    const float* k   = (const float*)d_in[1];  // (N,H,D)
    const float* val = (const float*)d_in[2];  // (N,H,M)
    const float* Si  = (const float*)d_in[3];  // (N,H,D,M)
    const float* Zi  = (const float*)d_in[4];  // (N,H,D)
    const float* Pi  = (const float*)d_in[5];  // (N,H,D,M)
    const float* Mi  = (const float*)d_in[6];  // (N,H,D,M)

    const size_t qsize = (size_t)in_sizes[0];  // N*H*D
    const size_t vsize = (size_t)in_sizes[2];  // N*H*M
    const size_t ssize = (size_t)in_sizes[3];  // N*H*D*M
    const int    NH    = (int)(qsize / 64);    // one block per (n,h)

    // Outputs concatenated in return order: V, Si_new, Zi_new, Pi_new, Mi_new, Ui
    float* out = (float*)d_out;
    float* oV  = out;
    float* oSi = oV  + vsize;
    float* oZi = oSi + ssize;
    float* oPi = oZi + qsize;
    float* oMi = oPi + ssize;
    float* oUi = oMi + ssize;

    ra_attn_kernel<<<NH, 256, 0, stream>>>(q, k, val, Si, Zi, Pi, Mi,
                                           oV, oSi, oZi, oPi, oMi, oUi);
}